// Attention_13073880449106
// MI455X (gfx1250) — compile-verified
//
#include <hip/hip_runtime.h>

typedef __attribute__((ext_vector_type(2))) float v2f;
typedef __attribute__((ext_vector_type(4))) float v4f;
typedef __attribute__((ext_vector_type(8))) float v8f;

#define SEQ 2048
#define BATCH 32
#define HID 1024

// -------------------------------------------------------------------------
// Kernel 1: q[m][n] = sum_k hidden[m][k] * W[k][n]   (M=32, N=1024, K=1024)
// One wave32 per 16x16 output tile using V_WMMA_F32_16X16X4_F32.
// A layout (16x4 f32): lanes 0-15 -> K={0,1} in VGPR{0,1}; lanes 16-31 -> K={2,3}
// B layout (4x16 f32): VGPR0 = row K=(0|2) across lanes, VGPR1 = row K=(1|3)
// D layout (16x16 f32): VGPR r: lanes 0-15 -> (M=r, N=lane); lanes 16-31 -> (M=r+8)
// -------------------------------------------------------------------------
__global__ __launch_bounds__(32)
void attn_qgemm_wmma(const float* __restrict__ hidden,
                     const float* __restrict__ W,
                     float* __restrict__ q) {
  const int lane = threadIdx.x & 31;
  const int n0   = blockIdx.x * 16;
  const int m0   = blockIdx.y * 16;
  const int half = lane >> 4;     // 0: lanes 0-15, 1: lanes 16-31
  const int l16  = lane & 15;
  const int kk   = half * 2;      // K sub-offset within the 4-wide slab

  v8f c = {};
  const float* arow = hidden + (size_t)(m0 + l16) * HID;
  for (int k0 = 0; k0 < HID; k0 += 4) {
    v2f a = *(const v2f*)(arow + k0 + kk);                 // A[M][kk], A[M][kk+1]
    v2f bf;
    bf.x = W[(size_t)(k0 + kk)     * HID + n0 + l16];      // B[kk][N]
    bf.y = W[(size_t)(k0 + kk + 1) * HID + n0 + l16];      // B[kk+1][N]
    c = __builtin_amdgcn_wmma_f32_16x16x4_f32(false, a, false, bf,
                                              (short)0, c, false, false);
  }
#pragma unroll
  for (int r = 0; r < 8; ++r) {
    const int m = m0 + r + half * 8;
    const int n = n0 + l16;
    q[(size_t)m * HID + n] = c[r];
  }
}

// -------------------------------------------------------------------------
// Kernel 2 (bandwidth-dominant): scores[b][s] = dot(q[b], enc[s][b])
// grid (SEQ/8, BATCH), 256 threads = 8 waves, one s per wave.
// enc is 256 MB streamed exactly once -> non-temporal loads.
// -------------------------------------------------------------------------
__global__ __launch_bounds__(256)
void attn_score(const float* __restrict__ enc,
                const float* __restrict__ q,
                float* __restrict__ scores) {
  __shared__ float qs[HID];
  const int b   = blockIdx.y;
  const int tid = threadIdx.x;

  // stage q[b][:] into LDS (256 threads x float4 = 1024 floats)
  ((v4f*)qs)[tid] = ((const v4f*)(q + (size_t)b * HID))[tid];
  __syncthreads();

  const int wave = tid >> 5;
  const int lane = tid & 31;
  const int s    = blockIdx.x * 8 + wave;
  const float* p = enc + ((size_t)s * BATCH + b) * HID;

  float acc = 0.f;
#pragma unroll
  for (int i = 0; i < 8; ++i) {
    const int idx = lane * 4 + i * 128;
    v4f v  = __builtin_nontemporal_load((const v4f*)(p + idx));  // TH=NT stream
    v4f qv = *(const v4f*)(qs + idx);
    acc += v.x * qv.x + v.y * qv.y + v.z * qv.z + v.w * qv.w;
  }
#pragma unroll
  for (int m = 16; m >= 1; m >>= 1)
    acc += __shfl_xor(acc, m, 32);          // wave32 reduction
  if (lane == 0) scores[(size_t)b * SEQ + s] = acc;
}

// -------------------------------------------------------------------------
// Kernel 3: out[b][s] = softmax_s(scores[b][s]); one block per batch row.
// (bias term h·b is constant over s -> softmax-invariant -> dropped)
// -------------------------------------------------------------------------
__global__ __launch_bounds__(256)
void attn_softmax(const float* __restrict__ scores,
                  float* __restrict__ out) {
  __shared__ float red[8];
  const int b    = blockIdx.x;
  const int tid  = threadIdx.x;
  const int wave = tid >> 5;
  const int lane = tid & 31;
  const float* row = scores + (size_t)b * SEQ;

  float v[8];
  float mx = -__builtin_inff();
#pragma unroll
  for (int i = 0; i < 8; ++i) {
    v[i] = row[tid + i * 256];
    mx = fmaxf(mx, v[i]);
  }
#pragma unroll
  for (int m = 16; m >= 1; m >>= 1)
    mx = fmaxf(mx, __shfl_xor(mx, m, 32));
  if (lane == 0) red[wave] = mx;
  __syncthreads();
  float bm = red[0];
#pragma unroll
  for (int w = 1; w < 8; ++w) bm = fmaxf(bm, red[w]);
  __syncthreads();

  float sum = 0.f;
#pragma unroll
  for (int i = 0; i < 8; ++i) {
    v[i] = __expf(v[i] - bm);
    sum += v[i];
  }
#pragma unroll
  for (int m = 16; m >= 1; m >>= 1)
    sum += __shfl_xor(sum, m, 32);
  if (lane == 0) red[wave] = sum;
  __syncthreads();
  float tot = 0.f;
#pragma unroll
  for (int w = 0; w < 8; ++w) tot += red[w];
  const float inv = 1.f / tot;
#pragma unroll
  for (int i = 0; i < 8; ++i)
    out[(size_t)b * SEQ + tid + i * 256] = v[i] * inv;
}

// -------------------------------------------------------------------------
extern "C" void kernel_launch(void* const* d_in, const int* in_sizes, int n_in,
                              void* d_out, int out_size, void* d_ws, size_t ws_size,
                              hipStream_t stream) {
  const float* hidden = (const float*)d_in[0];   // [1, 32, 1024]
  const float* enc    = (const float*)d_in[1];   // [2048, 32, 1024]
  const float* W      = (const float*)d_in[2];   // [1024, 1024]
  // d_in[3] = bias: zeros AND softmax-invariant -> unused
  float* out    = (float*)d_out;                 // [32, 1, 2048]
  float* q      = (float*)d_ws;                  // 32*1024 floats  (128 KB)
  float* scores = q + BATCH * HID;               // 32*2048 floats  (256 KB)

  attn_qgemm_wmma<<<dim3(HID / 16, BATCH / 16), 32, 0, stream>>>(hidden, W, q);
  attn_score<<<dim3(SEQ / 8, BATCH), 256, 0, stream>>>(enc, q, scores);
  attn_softmax<<<BATCH, 256, 0, stream>>>(scores, out);
}